// GraphSAGE_28973849378860
// MI455X (gfx1250) — compile-verified
//
#include <hip/hip_runtime.h>

typedef __attribute__((ext_vector_type(2))) float v2f;
typedef __attribute__((ext_vector_type(8))) float v8f;

#define NNODES 100000
#define NEDGES 1600000
#define DIM    128
#define LDSTR  132   // LDS row stride (dwords): conflict-free fragment reads, 16B-aligned rows
#define MT     2     // M tiles (of 16 rows) per block; 100000 % 32 == 0

static_assert(NNODES % (16 * MT) == 0, "no remainder tiles");
static_assert((LDSTR * 4) % 16 == 0, "LDS rows must stay 16B aligned");

// ---------------------------------------------------------------- zero (float4)
__global__ __launch_bounds__(256) void zero_f32v4(float4* __restrict__ p, int n4) {
    int i = blockIdx.x * 256 + threadIdx.x;
    if (i < n4) p[i] = float4{0.f, 0.f, 0.f, 0.f};
}

// ---------------------------------------------------------------- edge scatter
// one wave32 per edge: lane loads float4 (16B) of source row, atomic-adds into agg[dst]
__global__ __launch_bounds__(256) void edge_scatter(const float* __restrict__ h,
                                                    const int*   __restrict__ src,
                                                    const int*   __restrict__ dst,
                                                    float* __restrict__ agg,
                                                    float* __restrict__ deg) {
    int wave = threadIdx.x >> 5;
    int lane = threadIdx.x & 31;
    long e = (long)blockIdx.x * 8 + wave;
    if (e >= NEDGES) return;
    int s = src[e];
    int d = dst[e];
    const float4* hp = (const float4*)(h + (long)s * DIM);
    float4 v = hp[lane];                        // 32 lanes x 4 = 128 floats (L2-resident)
    float* ap = agg + (long)d * DIM + lane * 4;
    atomicAdd(ap + 0, v.x);
    atomicAdd(ap + 1, v.y);
    atomicAdd(ap + 2, v.z);
    atomicAdd(ap + 3, v.w);
    if (lane == 0) atomicAdd(deg + d, 1.0f);
}

// ---------------------------------------------------------------- fused dual-GEMM
// out[i,:] = act( h[i,:] @ Wself + (agg[i,:]/max(deg[i],1)) @ Wneigh + bias )
// block = 8 waves, owns 32 rows (2 M-tiles); wave w owns cols [16w,16w+16).
// Per K-step: load B fragments once, issue 2 WMMAs per matrix (one per M tile)
// -> 4 x V_WMMA_F32_16X16X4_F32 per K-step, 128 per wave.
template <bool RELU>
__global__ __launch_bounds__(256) void sage_gemm(const float* __restrict__ h,
                                                 const float* __restrict__ agg,
                                                 const float* __restrict__ deg,
                                                 const float* __restrict__ Wself,
                                                 const float* __restrict__ Wneigh,
                                                 const float* __restrict__ bias,
                                                 float* __restrict__ out) {
    __shared__ float shS[16 * MT * LDSTR];
    __shared__ float shM[16 * MT * LDSTR];

    const int rowBase = blockIdx.x * (16 * MT);

    // stage 32x128 tiles of h and mean into LDS with float4 transfers
    // (16*MT rows) * (DIM/4 chunks) = 32*32 = 1024 float4 per array
    for (int i = threadIdx.x; i < 16 * MT * (DIM / 4); i += 256) {
        int r  = i >> 5;          // DIM/4 == 32 chunks per row
        int c4 = i & 31;
        int g  = rowBase + r;     // always < NNODES (exact tiling)
        const float4 hs = *(const float4*)(h   + (long)g * DIM + c4 * 4);
        float4       hm = *(const float4*)(agg + (long)g * DIM + c4 * 4);
        const float rd = 1.0f / fmaxf(deg[g], 1.0f);
        hm.x *= rd; hm.y *= rd; hm.z *= rd; hm.w *= rd;
        *(float4*)(shS + r * LDSTR + c4 * 4) = hs;
        *(float4*)(shM + r * LDSTR + c4 * 4) = hm;
    }
    __syncthreads();

    const int wave = threadIdx.x >> 5;
    const int lane = threadIdx.x & 31;
    const int lr   = lane & 15;      // A: row M / B,D: col N
    const int hi   = lane >> 4;      // A: K pair select / D: M+8 select
    const int ncol = wave * 16 + lr;
    const int koff = hi * 2;

    v8f acc0 = {};
    v8f acc1 = {};
    #pragma unroll 4
    for (int k = 0; k < DIM; k += 4) {
        const int kk = k + koff;
        // B fragments (4x16 f32): lane<16 -> rows K,K+1 ; lane>=16 -> K+2,K+3 at col ncol
        v2f bS; bS.x = Wself [kk * DIM + ncol]; bS.y = Wself [(kk + 1) * DIM + ncol];
        v2f bM; bM.x = Wneigh[kk * DIM + ncol]; bM.y = Wneigh[(kk + 1) * DIM + ncol];
        // A fragments (16x4 f32): lane<16 -> K,K+1 ; lane>=16 -> K+2,K+3 of row lr
        v2f aS0; aS0.x = shS[ lr       * LDSTR + kk]; aS0.y = shS[ lr       * LDSTR + kk + 1];
        v2f aM0; aM0.x = shM[ lr       * LDSTR + kk]; aM0.y = shM[ lr       * LDSTR + kk + 1];
        v2f aS1; aS1.x = shS[(lr + 16) * LDSTR + kk]; aS1.y = shS[(lr + 16) * LDSTR + kk + 1];
        v2f aM1; aM1.x = shM[(lr + 16) * LDSTR + kk]; aM1.y = shM[(lr + 16) * LDSTR + kk + 1];

        acc0 = __builtin_amdgcn_wmma_f32_16x16x4_f32(false, aS0, false, bS, (short)0, acc0, false, false);
        acc1 = __builtin_amdgcn_wmma_f32_16x16x4_f32(false, aS1, false, bS, (short)0, acc1, false, false);
        acc0 = __builtin_amdgcn_wmma_f32_16x16x4_f32(false, aM0, false, bM, (short)0, acc0, false, false);
        acc1 = __builtin_amdgcn_wmma_f32_16x16x4_f32(false, aM1, false, bM, (short)0, acc1, false, false);
    }

    // epilogue: D layout -> VGPR r holds row (r + 8*hi), col ncol. No bounds checks needed.
    const float bv = bias[ncol];
    float* outBase = out + (long)(rowBase + hi * 8) * DIM + ncol;
    #pragma unroll
    for (int r = 0; r < 8; ++r) {
        float v0 = acc0[r] + bv;
        float v1 = acc1[r] + bv;
        if (RELU) { v0 = fmaxf(v0, 0.0f); v1 = fmaxf(v1, 0.0f); }
        outBase[(long)r * DIM]        = v0;
        outBase[(long)(r + 16) * DIM] = v1;
    }
}

// ---------------------------------------------------------------- launcher
extern "C" void kernel_launch(void* const* d_in, const int* in_sizes, int n_in,
                              void* d_out, int out_size, void* d_ws, size_t ws_size,
                              hipStream_t stream) {
    (void)in_sizes; (void)n_in; (void)out_size; (void)ws_size;

    const float* x   = (const float*)d_in[0];
    const int*   src = (const int*)  d_in[1];
    const int*   dst = (const int*)  d_in[2];
    const float* Ws[4]; const float* Wn[4]; const float* B[4];
    for (int l = 0; l < 4; ++l) {
        Ws[l] = (const float*)d_in[3 + 3 * l];
        Wn[l] = (const float*)d_in[4 + 3 * l];
        B[l]  = (const float*)d_in[5 + 3 * l];
    }

    float* agg  = (float*)d_ws;                    // N*D
    float* deg  = agg + (size_t)NNODES * DIM;      // N   (contiguous after agg)
    float* hbuf = deg + NNODES;                    // N*D
    float* outf = (float*)d_out;

    const int ztot4 = (NNODES * DIM + NNODES) / 4; // agg + deg zeroed together (both %4==0)
    const int zgrid = (ztot4 + 255) / 256;
    const int egrid = (NEDGES + 7) / 8;
    const int ggrid = NNODES / (16 * MT);          // exact

    const float* hin = x;
    float* houts[4] = { hbuf, outf, hbuf, outf };

    for (int l = 0; l < 4; ++l) {
        zero_f32v4<<<zgrid, 256, 0, stream>>>((float4*)agg, ztot4);
        edge_scatter<<<egrid, 256, 0, stream>>>(hin, src, dst, agg, deg);
        if (l < 3)
            sage_gemm<true ><<<ggrid, 256, 0, stream>>>(hin, agg, deg, Ws[l], Wn[l], B[l], houts[l]);
        else
            sage_gemm<false><<<ggrid, 256, 0, stream>>>(hin, agg, deg, Ws[l], Wn[l], B[l], houts[l]);
        hin = houts[l];
    }
}